// NoisyTopKRouter_68358699483591
// MI455X (gfx1250) — compile-verified
//
#include <hip/hip_runtime.h>
#include <hip/hip_bf16.h>

// ---------------------------------------------------------------------------
// NoisyTopKRouter for MI455X (gfx1250, wave32, WMMA + async LDS copies)
//   B=131072, D=1024, E=64, K=2
//   d_out layout (flat, fp32): gates[B*E] | load[E] | logits[B*E] | lb[1] | importance[E]
// ---------------------------------------------------------------------------

#define B_ROWS   131072
#define D_DIM    1024
#define E_DIM    64
#define KC       64            // K-chunk staged in LDS
#define XP       (KC + 4)      // padded x row stride in LDS (bank-conflict pad)
#define ROWS_PB  128           // rows per block (8 waves x 16 rows)
#define NKC      (D_DIM / KC)  // 16 chunks
#define BUFSZ    16896         // floats per LDS buffer: 4096 + 4096 + 128*68
#define BE       ((size_t)B_ROWS * E_DIM)

typedef __attribute__((ext_vector_type(2))) float v2f;
typedef __attribute__((ext_vector_type(8))) float v8f;

__device__ __forceinline__ float softplus_f(float x) {
    return (x > 20.0f) ? x : log1pf(expf(x));
}
__device__ __forceinline__ float ndtr_f(float x) {
    return 0.5f * erfcf(-x * 0.70710678118654752f);
}

// Async global->LDS 128-bit copy (GVS addressing: 64-bit SGPR base + 32-bit
// per-lane unsigned offset). Tracked by ASYNCcnt. LDS dest address = low 32
// bits of the generic pointer.
__device__ __forceinline__ void async_ld128(void* lds, uint64_t sbase, uint32_t voff) {
    uint32_t lo = (uint32_t)(uintptr_t)lds;
    asm volatile("global_load_async_to_lds_b128 %0, %1, %2"
                 :: "v"(lo), "v"(voff), "s"(sbase)
                 : "memory");
}
__device__ __forceinline__ void wait_async0() {
    asm volatile("s_wait_asynccnt 0x0" ::: "memory");
}

// ---------------------------------------------------------------------------
__global__ void zero_ws_kernel(float* ws) {
    int t = blockIdx.x * blockDim.x + threadIdx.x;
    if (t < 129) ws[t] = 0.0f;     // load[64] | importance[64] | z_sum[1]
}

// ---------------------------------------------------------------------------
// Main fused kernel: dual GEMM (WMMA f32 16x16x4) + router epilogue.
// Dynamic LDS: two staging buffers (double-buffered async copies):
//   buf b at smem + b*BUFSZ:  sWg[4096] | sWn[4096] | sX[128*68]
// Epilogue (overlaps buf0):   sC[128*64] | sN[128*64]
// ---------------------------------------------------------------------------
__global__ __launch_bounds__(256) void router_main_kernel(
    const float* __restrict__ x, const float* __restrict__ wg,
    const float* __restrict__ wn, const float* __restrict__ eps,
    float* __restrict__ out, float* __restrict__ ws)
{
    extern __shared__ float smem[];

    __shared__ float loadP[E_DIM];
    __shared__ float impP[E_DIM];
    __shared__ float zP;

    const int tid     = threadIdx.x;
    const int lane    = tid & 31;
    const int wave    = tid >> 5;          // 0..7
    const int hi      = lane >> 4;         // half-wave select
    const int ln      = lane & 15;
    const int rowBase = blockIdx.x * ROWS_PB;

    if (tid < E_DIM) { loadP[tid] = 0.0f; impP[tid] = 0.0f; }
    if (tid == 0) zP = 0.0f;

    // ---- async stage of one K-chunk into LDS buffer `buf` ----
    auto stage_chunk = [&](int kc, int buf) {
        float* bWg = smem + buf * BUFSZ;
        float* bWn = bWg + 4096;
        float* bX  = bWn + 4096;
        uint64_t wgB = (uint64_t)(uintptr_t)(wg + (size_t)kc * KC * E_DIM);
        uint64_t wnB = (uint64_t)(uintptr_t)(wn + (size_t)kc * KC * E_DIM);
        uint64_t xB  = (uint64_t)(uintptr_t)(x + (size_t)rowBase * D_DIM + kc * KC);
        #pragma unroll
        for (int i = 0; i < 4; ++i) {
            int idx = tid + i * 256;                  // 0..1023 float4
            uint32_t vo = (uint32_t)idx * 16u;
            async_ld128(&bWg[idx * 4], wgB, vo);
            async_ld128(&bWn[idx * 4], wnB, vo);
        }
        #pragma unroll
        for (int i = 0; i < 8; ++i) {
            int idx = tid + i * 256;                  // 0..2047 float4
            int r = idx >> 4, c4 = idx & 15;
            async_ld128(&bX[r * XP + c4 * 4], xB,
                        (uint32_t)(r * D_DIM * 4 + c4 * 16));
        }
    };

    v8f accg[4], accn[4];
    #pragma unroll
    for (int nt = 0; nt < 4; ++nt) { accg[nt] = (v8f){}; accn[nt] = (v8f){}; }

    stage_chunk(0, 0);

    for (int kc = 0; kc < NKC; ++kc) {
        const int cur = kc & 1;
        float* cWg = smem + cur * BUFSZ;
        float* cWn = cWg + 4096;
        float* cX  = cWn + 4096;

        wait_async0();          // chunk kc landed in LDS (per-wave)
        __syncthreads();        // publish across workgroup

        if (kc + 1 < NKC) stage_chunk(kc + 1, cur ^ 1);   // overlap next copy

        // ---- WMMA: 16 K-steps of K=4; 4 N-tiles x {gate,noise} ----
        for (int k4 = 0; k4 < KC / 4; ++k4) {
            const int k0 = k4 * 4 + 2 * hi;
            // A fragment: 16x4 f32 -> 2 VGPRs; lane row = ln, K = k0, k0+1
            v2f a = *(const v2f*)&cX[(wave * 16 + ln) * XP + k0];
            const int bbase = k0 * E_DIM + ln;
            #pragma unroll
            for (int nt = 0; nt < 4; ++nt) {
                v2f bg, bn;
                bg.x = cWg[bbase + nt * 16];
                bg.y = cWg[bbase + nt * 16 + E_DIM];
                accg[nt] = __builtin_amdgcn_wmma_f32_16x16x4_f32(
                    false, a, false, bg, (short)0, accg[nt], false, false);
                bn.x = cWn[bbase + nt * 16];
                bn.y = cWn[bbase + nt * 16 + E_DIM];
                accn[nt] = __builtin_amdgcn_wmma_f32_16x16x4_f32(
                    false, a, false, bn, (short)0, accn[nt], false, false);
            }
        }
        __syncthreads();        // done reading `cur` before it is overwritten
    }

    // ---- spill accumulators to LDS (reuse buffer 0 region) ----
    float* sC = smem;          // ROWS_PB * 64
    float* sN = smem + 8192;   // ROWS_PB * 64
    #pragma unroll
    for (int nt = 0; nt < 4; ++nt) {
        #pragma unroll
        for (int i = 0; i < 8; ++i) {
            int r = wave * 16 + i + 8 * hi;
            int c = ln + nt * 16;
            sC[r * E_DIM + c] = accg[nt][i];
            sN[r * E_DIM + c] = accn[nt][i];
        }
    }
    __syncthreads();

    // ---- per-row router epilogue: one thread per row ----
    float* gates  = out;                       // [B, E]
    float* logitO = out + BE + E_DIM;          // [B, E]

    if (tid < ROWS_PB) {
        const int row = rowBase + tid;
        const float* erow = eps + (size_t)row * E_DIM;
        float* grow = gates  + (size_t)row * E_DIM;
        float* lrow = logitO + (size_t)row * E_DIM;

        // pass 1: logits + top-3 (values) + top-2 (indices)
        float v0 = -1e30f, v1 = -1e30f, v2 = -1e30f;
        int   i0 = 0, i1 = 0;
        #pragma unroll 4
        for (int e = 0; e < E_DIM; ++e) {
            float c  = sC[tid * E_DIM + e];
            float sd = softplus_f(sN[tid * E_DIM + e]) + 0.01f;
            float l  = c + erow[e] * sd;
            lrow[e]  = l;
            if (l > v0)      { v2 = v1; v1 = v0; i1 = i0; v0 = l; i0 = e; }
            else if (l > v1) { v2 = v1; v1 = l; i1 = e; }
            else if (l > v2) { v2 = l; }
        }
        // softmax over top-2
        float g0 = 1.0f / (1.0f + expf(v1 - v0));
        float g1 = 1.0f - g0;
        // scatter gates (zero row first, then the two winners)
        float4 z4 = {0.0f, 0.0f, 0.0f, 0.0f};
        #pragma unroll
        for (int e4 = 0; e4 < E_DIM / 4; ++e4) ((float4*)grow)[e4] = z4;
        grow[i0] = g0;
        grow[i1] = g1;
        atomicAdd(&impP[i0], g0);
        atomicAdd(&impP[i1], g1);

        // pass 2: logsumexp + load (ndtr probabilities vs thresholds)
        //   thr_if_in = 3rd largest (v2), thr_if_out = 2nd largest (v1)
        float se = 0.0f;
        #pragma unroll 4
        for (int e = 0; e < E_DIM; ++e) {
            float c  = sC[tid * E_DIM + e];
            float sd = softplus_f(sN[tid * E_DIM + e]) + 0.01f;
            float l  = c + erow[e] * sd;
            se += expf(l - v0);
            float thr = (l > v2) ? v2 : v1;
            atomicAdd(&loadP[e], ndtr_f((c - thr) / sd));
        }
        atomicAdd(&zP, v0 + logf(se));
    }
    __syncthreads();

    // ---- flush block partials to global workspace ----
    if (tid < E_DIM) {
        atomicAdd(&ws[tid],          loadP[tid]);   // load
        atomicAdd(&ws[E_DIM + tid],  impP[tid]);    // importance
    }
    if (tid == 0) atomicAdd(&ws[2 * E_DIM], zP);    // z sum
}

// ---------------------------------------------------------------------------
__global__ void finalize_kernel(const float* __restrict__ ws, float* __restrict__ out) {
    int t = threadIdx.x;       // 64 threads
    float ld  = ws[t];
    float imp = ws[E_DIM + t];
    out[BE + t]                 = ld;    // load
    out[2 * BE + E_DIM + 1 + t] = imp;   // importance
    __syncthreads();
    if (t == 0) {
        float ms = 0.0f, mi = 0.0f;
        for (int e = 0; e < E_DIM; ++e) { ms += ws[e]; mi += ws[E_DIM + e]; }
        ms /= (float)E_DIM; mi /= (float)E_DIM;
        float vs = 0.0f, vi = 0.0f;
        for (int e = 0; e < E_DIM; ++e) {
            float a = ws[e] - ms;          vs += a * a;
            float b = ws[E_DIM + e] - mi;  vi += b * b;
        }
        vs /= (float)(E_DIM - 1); vi /= (float)(E_DIM - 1);
        float cv_load = vs / (ms * ms + 1e-10f);
        float cv_imp  = vi / (mi * mi + 1e-10f);
        float z_loss  = ws[2 * E_DIM] / (float)B_ROWS;
        out[2 * BE + E_DIM] = cv_imp + cv_load + z_loss;   // load_balancing
    }
}

// ---------------------------------------------------------------------------
extern "C" void kernel_launch(void* const* d_in, const int* in_sizes, int n_in,
                              void* d_out, int out_size, void* d_ws, size_t ws_size,
                              hipStream_t stream) {
    const float* x   = (const float*)d_in[0];
    const float* wg  = (const float*)d_in[1];
    const float* wn  = (const float*)d_in[2];
    const float* eps = (const float*)d_in[3];
    float* out = (float*)d_out;
    float* ws  = (float*)d_ws;

    zero_ws_kernel<<<1, 256, 0, stream>>>(ws);

    const int    nblocks = B_ROWS / ROWS_PB;          // 1024
    const size_t smem    = 2 * BUFSZ * sizeof(float); // 135168 B (double buffer)
    router_main_kernel<<<nblocks, 256, smem, stream>>>(x, wg, wn, eps, out, ws);

    finalize_kernel<<<1, 64, 0, stream>>>(ws, out);
}